// Encoder_14748917694972
// MI455X (gfx1250) — compile-verified
//
#include <hip/hip_runtime.h>
#include <hip/hip_bf16.h>

typedef float v2f __attribute__((ext_vector_type(2)));
typedef float v8f __attribute__((ext_vector_type(8)));

#define NNODES 50000
#define CIN    128
#define HID1   512
#define HID2   64

// ---------------------------------------------------------------------------
// Degree / normalization
// ---------------------------------------------------------------------------
__global__ void k_deg_init(float* deg, int n) {
    int i = blockIdx.x * blockDim.x + threadIdx.x;
    if (i < n) deg[i] = 1.0f;                 // self loop contributes 1
}

__global__ void k_deg_count(const int* __restrict__ dst, float* deg, int e) {
    int i = blockIdx.x * blockDim.x + threadIdx.x;
    if (i < e) atomicAdd(&deg[dst[i]], 1.0f);
}

__global__ void k_dinv(float* deg, int n) {
    int i = blockIdx.x * blockDim.x + threadIdx.x;
    if (i < n) {
        float d = deg[i];
        deg[i] = (d > 0.0f) ? rsqrtf(d) : 0.0f;   // overwrite in place -> dinv
    }
}

// ---------------------------------------------------------------------------
// fp32 WMMA GEMM:  C[M x NOUT] = A[M x K] * B[K x NOUT]
// Block = 128 threads (4 waves). Each wave owns NACC adjacent 16x16 tiles
// (16 x 16*NACC of C): every A fragment feeds NACC back-to-back WMMAs.
// B fragments use a copy-free ping-pong (bf0/bf1), refilled in place right
// after the WMMA group that consumes them: a hand-made 2-stage pipeline.
// The compiler's MachinePipeliner/unroller is disabled on the loop so it
// does not add a rotation stage (acc register shuffling movs).
// grid.x = M/16 row tiles, grid.y = NOUT/(64*NACC) column groups.
// FUSE: apply prelu(v + pb[c], pa[c]) while staging the A tile (c = K index).
// ---------------------------------------------------------------------------
template <int K, int NOUT, int NACC, bool FUSE>
__global__ __launch_bounds__(128) void k_gemm_wmma_f32(
    const float* __restrict__ A, const float* __restrict__ B,
    float* __restrict__ C,
    const float* __restrict__ pb, const float* __restrict__ pa) {
    constexpr int LDA = K + 4;                 // 16B-aligned rows, conflict-free frags
    __shared__ float sA[16 * LDA];

    const int row0 = blockIdx.x * 16;
    const int wave = threadIdx.x >> 5;         // 0..3
    const int lane = threadIdx.x & 31;
    const int col0 = blockIdx.y * (64 * NACC) + wave * (16 * NACC);
    const int half = lane >> 4;                // 0 or 1
    const int l16  = lane & 15;

    // Cooperative float4 staging of the 16 x K A-tile (optionally fused prelu)
    for (int i = threadIdx.x; i < 16 * (K / 4); i += 128) {
        int r = i / (K / 4), c4 = (i % (K / 4)) * 4;
        float4 v = *(const float4*)&A[(size_t)(row0 + r) * K + c4];
        if constexpr (FUSE) {
            float t0 = v.x + pb[c4 + 0]; v.x = (t0 >= 0.f) ? t0 : pa[c4 + 0] * t0;
            float t1 = v.y + pb[c4 + 1]; v.y = (t1 >= 0.f) ? t1 : pa[c4 + 1] * t1;
            float t2 = v.z + pb[c4 + 2]; v.z = (t2 >= 0.f) ? t2 : pa[c4 + 2] * t2;
            float t3 = v.w + pb[c4 + 3]; v.w = (t3 >= 0.f) ? t3 : pa[c4 + 3] * t3;
        }
        *(float4*)&sA[r * LDA + c4] = v;
    }
    __syncthreads();

    v8f acc[NACC];
#pragma unroll
    for (int j = 0; j < NACC; ++j) acc[j] = (v8f){};

    // Per-lane bases. A frag at step k: ap[k], ap[k+1].
    // B frag at step k: row k + half*2 (+1 for .y), col col0 + l16 + j*16.
    const float* ap   = &sA[l16 * LDA + half * 2];
    const float* base = &B[(size_t)(half * 2) * NOUT + col0 + l16];

    // Prologue: fill both ping-pong buffers (k = 0 and k = 4)
    v2f bf0[NACC], bf1[NACC];
#pragma unroll
    for (int j = 0; j < NACC; ++j) {
        bf0[j].x = base[j * 16];
        bf0[j].y = base[j * 16 + NOUT];
        bf1[j].x = base[4 * NOUT + j * 16];
        bf1[j].y = base[4 * NOUT + j * 16 + NOUT];
    }
    const float* bp = base + 8 * NOUT;         // next refill: rows for k+8

#pragma clang loop pipeline(disable) unroll(disable)
    for (int k = 0; k < K - 8; k += 8) {
        // ---- step k: consume bf0, then refill it with step k+8 ----
        v2f a0; a0.x = ap[k]; a0.y = ap[k + 1];
#pragma unroll
        for (int j = 0; j < NACC; ++j)
            acc[j] = __builtin_amdgcn_wmma_f32_16x16x4_f32(
                false, a0, false, bf0[j], (short)0, acc[j], false, false);
#pragma unroll
        for (int j = 0; j < NACC; ++j) {
            bf0[j].x = bp[j * 16];
            bf0[j].y = bp[j * 16 + NOUT];
        }
        // ---- step k+4: consume bf1, then refill it with step k+12 ----
        v2f a1; a1.x = ap[k + 4]; a1.y = ap[k + 5];
#pragma unroll
        for (int j = 0; j < NACC; ++j)
            acc[j] = __builtin_amdgcn_wmma_f32_16x16x4_f32(
                false, a1, false, bf1[j], (short)0, acc[j], false, false);
#pragma unroll
        for (int j = 0; j < NACC; ++j) {
            bf1[j].x = bp[4 * NOUT + j * 16];
            bf1[j].y = bp[4 * NOUT + j * 16 + NOUT];
        }
        bp += 8 * NOUT;
    }
    {   // Tail: steps K-8 (bf0) and K-4 (bf1), no refills
        v2f a0; a0.x = ap[K - 8]; a0.y = ap[K - 7];
#pragma unroll
        for (int j = 0; j < NACC; ++j)
            acc[j] = __builtin_amdgcn_wmma_f32_16x16x4_f32(
                false, a0, false, bf0[j], (short)0, acc[j], false, false);
        v2f a1; a1.x = ap[K - 4]; a1.y = ap[K - 3];
#pragma unroll
        for (int j = 0; j < NACC; ++j)
            acc[j] = __builtin_amdgcn_wmma_f32_16x16x4_f32(
                false, a1, false, bf1[j], (short)0, acc[j], false, false);
    }

    // C/D layout: VGPR r -> row = r + 8*half, col = l16
    float* cp = C + (size_t)(row0 + half * 8) * NOUT + col0 + l16;
#pragma unroll
    for (int j = 0; j < NACC; ++j)
#pragma unroll
        for (int r = 0; r < 8; ++r)
            cp[(size_t)r * NOUT + j * 16] = acc[j][r];
}

// ---------------------------------------------------------------------------
// Aggregation: init with self-loop term, then edge-parallel atomic scatter
// ---------------------------------------------------------------------------
template <int C>
__global__ void k_init_self(const float4* __restrict__ h,
                            const float* __restrict__ dinv,
                            float4* __restrict__ out, int n) {
    int i = blockIdx.x * blockDim.x + threadIdx.x;
    if (i < n * (C / 4)) {
        float di = dinv[i / (C / 4)];
        float s = di * di;                     // self loop: norm = dinv^2
        float4 v = h[i];
        v.x *= s; v.y *= s; v.z *= s; v.w *= s;
        out[i] = v;
    }
}

__global__ __launch_bounds__(128) void k_scatter512(
    const int* __restrict__ src, const int* __restrict__ dst,
    const float* __restrict__ dinv, const float* __restrict__ h,
    float* __restrict__ out) {
    int e = blockIdx.x;
    int s = src[e], d = dst[e];
    float nrm = dinv[s] * dinv[d];
    int c4 = threadIdx.x * 4;                  // 128 threads x float4 = 512 ch
    float4 v = *(const float4*)(h + (size_t)s * HID1 + c4);
    float* op = out + (size_t)d * HID1 + c4;
    atomicAdd(op + 0, v.x * nrm);
    atomicAdd(op + 1, v.y * nrm);
    atomicAdd(op + 2, v.z * nrm);
    atomicAdd(op + 3, v.w * nrm);
}

__global__ __launch_bounds__(64) void k_scatter64(
    const int* __restrict__ src, const int* __restrict__ dst,
    const float* __restrict__ dinv, const float* __restrict__ h,
    float* __restrict__ out, int E) {
    int e = blockIdx.x * 4 + (threadIdx.x >> 4);   // 4 edges per block
    if (e >= E) return;
    int s = src[e], d = dst[e];
    float nrm = dinv[s] * dinv[d];
    int c4 = (threadIdx.x & 15) * 4;               // 16 threads x float4 = 64 ch
    float4 v = *(const float4*)(h + (size_t)s * HID2 + c4);
    float* op = out + (size_t)d * HID2 + c4;
    atomicAdd(op + 0, v.x * nrm);
    atomicAdd(op + 1, v.y * nrm);
    atomicAdd(op + 2, v.z * nrm);
    atomicAdd(op + 3, v.w * nrm);
}

// ---------------------------------------------------------------------------
// x = prelu(x + b, a)   (float4, C divisible by 4)
// ---------------------------------------------------------------------------
template <int C>
__global__ void k_bias_prelu(float4* __restrict__ x,
                             const float* __restrict__ b,
                             const float* __restrict__ a, int n) {
    int i = blockIdx.x * blockDim.x + threadIdx.x;
    if (i < n * (C / 4)) {
        int c = (i % (C / 4)) * 4;
        float4 v = x[i];
        float t0 = v.x + b[c + 0]; v.x = (t0 >= 0.f) ? t0 : a[c + 0] * t0;
        float t1 = v.y + b[c + 1]; v.y = (t1 >= 0.f) ? t1 : a[c + 1] * t1;
        float t2 = v.z + b[c + 2]; v.z = (t2 >= 0.f) ? t2 : a[c + 2] * t2;
        float t3 = v.w + b[c + 3]; v.w = (t3 >= 0.f) ? t3 : a[c + 3] * t3;
        x[i] = v;
    }
}

// ---------------------------------------------------------------------------
extern "C" void kernel_launch(void* const* d_in, const int* in_sizes, int n_in,
                              void* d_out, int out_size, void* d_ws, size_t ws_size,
                              hipStream_t stream) {
    const float* x   = (const float*)d_in[0];
    const int*   ei  = (const int*)d_in[1];
    const float* W1  = (const float*)d_in[2];
    const float* b1  = (const float*)d_in[3];
    const float* a1  = (const float*)d_in[4];
    const float* W2  = (const float*)d_in[5];
    const float* b2  = (const float*)d_in[6];
    const float* a2  = (const float*)d_in[7];
    float* out = (float*)d_out;

    const int E = in_sizes[1] / 2;
    const int* src = ei;
    const int* dst = ei + E;

    // Workspace layout (256B aligned):
    //   dinv : N floats
    //   h1   : N*512 floats      (later reused for h2 : N*64)
    //   agg1 : N*512 floats
    char* ws = (char*)d_ws;
    size_t off = 0;
    float* dinv = (float*)(ws + off); off += ((size_t)NNODES * 4 + 255) & ~(size_t)255;
    float* h1   = (float*)(ws + off); off += ((size_t)NNODES * HID1 * 4 + 255) & ~(size_t)255;
    float* agg1 = (float*)(ws + off);
    float* h2   = h1;   // h1 is dead once agg1 is finalized

    const int T = 256;

    // --- degrees -> dinv ---
    k_deg_init<<<(NNODES + T - 1) / T, T, 0, stream>>>(dinv, NNODES);
    k_deg_count<<<(E + T - 1) / T, T, 0, stream>>>(dst, dinv, E);
    k_dinv<<<(NNODES + T - 1) / T, T, 0, stream>>>(dinv, NNODES);

    // --- layer 1: h1 = x @ W1 (fp32 WMMA, 4 acc tiles/wave, ping-pong B) ---
    {
        dim3 grid(NNODES / 16, HID1 / (64 * 4));
        k_gemm_wmma_f32<CIN, HID1, 4, false><<<grid, 128, 0, stream>>>(
            x, W1, h1, nullptr, nullptr);
    }
    // aggregate: agg1 = self + scatter(edges)   (bias+prelu fused into GEMM2)
    {
        int tot4 = NNODES * (HID1 / 4);
        k_init_self<HID1><<<(tot4 + T - 1) / T, T, 0, stream>>>(
            (const float4*)h1, dinv, (float4*)agg1, NNODES);
        k_scatter512<<<E, 128, 0, stream>>>(src, dst, dinv, h1, agg1);
    }

    // --- layer 2: h2 = prelu(agg1 + b1, a1) @ W2 (prelu fused in A-stage) ---
    {
        dim3 grid(NNODES / 16, HID2 / 64);
        k_gemm_wmma_f32<HID1, HID2, 1, true><<<grid, 128, 0, stream>>>(
            agg1, W2, h2, b1, a1);
    }
    // aggregate into d_out, then final bias+prelu
    {
        int tot4 = NNODES * (HID2 / 4);
        k_init_self<HID2><<<(tot4 + T - 1) / T, T, 0, stream>>>(
            (const float4*)h2, dinv, (float4*)out, NNODES);
        k_scatter64<<<(E + 3) / 4, 64, 0, stream>>>(src, dst, dinv, h2, out, E);
        k_bias_prelu<HID2><<<(tot4 + T - 1) / T, T, 0, stream>>>(
            (float4*)out, b2, a2, NNODES);
    }
}